// PositionEmbedding_13030930776138
// MI455X (gfx1250) — compile-verified
//
#include <hip/hip_runtime.h>
#include <cstdint>
#include <cstddef>

// Problem constants from the reference: B=32, S=4096, H=512, V=8192, P=4096.
#define S_LEN 4096
#define H_DIM 512
#define THREADS 256
#define WAVES_PER_BLOCK 8   // 256 threads / wave32

typedef float v4f __attribute__((ext_vector_type(4)));

// ---------------------------------------------------------------------------
// Phase 1: per-row segmented scan.
//   starts[j]       = (j>0) && (state[j-1] == next_id)
//   instructions[j] = inclusive cumsum(starts)
//   arguments[j]    = j - cummax(starts[j] ? j : 0)
// One block per batch row; (sum, max) is a joint associative scan.
// ---------------------------------------------------------------------------
__global__ __launch_bounds__(256) void scan_kernel(const int* __restrict__ state,
                                                   const int* __restrict__ next_id_p,
                                                   int2* __restrict__ idx_out) {
  const int b = blockIdx.x;
  const int t = threadIdx.x;
  const int next_id = *next_id_p;
  constexpr int CH = S_LEN / THREADS;  // 16 elements per thread
  const int base = t * CH;
  const int* row = state + (size_t)b * S_LEN;

  int lcnt[CH], llast[CH];
  int cnt = 0, last = 0;
#pragma unroll
  for (int i = 0; i < CH; ++i) {
    const int j = base + i;
    const bool start = (j > 0) && (row[j - 1] == next_id);
    cnt += start ? 1 : 0;
    if (start) last = j;
    lcnt[i] = cnt;
    llast[i] = last;
  }

  __shared__ int scnt[THREADS];
  __shared__ int slast[THREADS];
  scnt[t] = cnt;
  slast[t] = last;
  __syncthreads();
  // Hillis-Steele inclusive scan over per-thread partials (sum, max).
  for (int off = 1; off < THREADS; off <<= 1) {
    int c = 0, l = 0;
    if (t >= off) { c = scnt[t - off]; l = slast[t - off]; }
    __syncthreads();
    scnt[t] += c;
    slast[t] = max(slast[t], l);
    __syncthreads();
  }
  const int exCnt  = (t > 0) ? scnt[t - 1] : 0;
  const int exLast = (t > 0) ? slast[t - 1] : 0;

  int2* orow = idx_out + (size_t)b * S_LEN;
#pragma unroll
  for (int i = 0; i < CH; ++i) {
    const int j = base + i;
    const int instr = exCnt + lcnt[i];           // < P == 4096 always
    const int lst   = max(exLast, llast[i]);
    orow[j] = make_int2(instr, j - lst);         // arg < 4096 always
  }
}

// ---------------------------------------------------------------------------
// Phase 2: one wave32 per (b,s) row.
//   - 12x global_load_async_to_lds_b128 stage the three gathered H=512 rows
//     (6 KB/wave) via the CDNA5 async data mover (ASYNCcnt).
//   - s_wait_asynccnt 0, sum rows, wave32 shuffle reduce mean/var,
//     normalize, scale/bias from LDS-cached ln params.
//   - Output is a 256 MB write-once stream: store with NT temporal hint so it
//     does not evict the ~32 MB of gather tables resident in the 192 MB L2.
// ---------------------------------------------------------------------------
__global__ __launch_bounds__(256) void emb_ln_kernel(const int* __restrict__ state,
                                                     const int2* __restrict__ idx,
                                                     const float* __restrict__ tok,
                                                     const float* __restrict__ ins,
                                                     const float* __restrict__ argt,
                                                     const float* __restrict__ w,
                                                     const float* __restrict__ bi,
                                                     float* __restrict__ out,
                                                     int nrows) {
  __shared__ __align__(16) float swb[2 * H_DIM];                       // 4 KB
  __shared__ __align__(16) float stage[WAVES_PER_BLOCK][3 * H_DIM];    // 48 KB

  const int tid = threadIdx.x;
  for (int i = tid; i < 2 * H_DIM; i += THREADS)
    swb[i] = (i < H_DIM) ? w[i] : bi[i - H_DIM];
  __syncthreads();

  const int wave = tid >> 5;
  const int lane = tid & 31;
  const long long row = (long long)blockIdx.x * WAVES_PER_BLOCK + wave;
  if (row >= nrows) return;

  const int tokId = state[row];
  const int2 ia = idx[row];
  const float* gs0 = tok  + (size_t)tokId * H_DIM;
  const float* gs1 = ins  + (size_t)ia.x  * H_DIM;
  const float* gs2 = argt + (size_t)ia.y  * H_DIM;
  const float* gs[3] = {gs0, gs1, gs2};

  float* sb = &stage[wave][0];

  // Async gather: lane covers 16 B; 4 chunks of 512 B cover one 2 KB row.
#pragma unroll
  for (int r = 0; r < 3; ++r) {
    const float* gbase = gs[r] + lane * 4;
    // low 32 bits of the generic LDS pointer == workgroup-relative LDS byte offset
    const uint32_t lbase =
        (uint32_t)(uintptr_t)(sb + r * H_DIM) + (uint32_t)lane * 16u;
#pragma unroll
    for (int c = 0; c < 4; ++c) {
      const float* g = gbase + c * 128;          // +512 B per chunk
      const uint32_t l = lbase + (uint32_t)c * 512u;
      asm volatile("global_load_async_to_lds_b128 %0, %1, off"
                   :
                   : "v"(l), "v"(g)
                   : "memory");
    }
  }
  asm volatile("s_wait_asynccnt 0" ::: "memory");

  // Sum the three rows; lane holds 16 floats at e = lane*4 + c*128.
  float v[16];
  float s = 0.f, sq = 0.f;
#pragma unroll
  for (int c = 0; c < 4; ++c) {
    const int e = lane * 4 + c * 128;
    const v4f a  = *(const v4f*)(sb + 0 * H_DIM + e);
    const v4f b4 = *(const v4f*)(sb + 1 * H_DIM + e);
    const v4f c4 = *(const v4f*)(sb + 2 * H_DIM + e);
    const v4f t = a + b4 + c4;
    v[c * 4 + 0] = t.x; v[c * 4 + 1] = t.y;
    v[c * 4 + 2] = t.z; v[c * 4 + 3] = t.w;
    s  += t.x + t.y + t.z + t.w;
    sq += t.x * t.x + t.y * t.y + t.z * t.z + t.w * t.w;
  }

  // wave32 butterfly reduction (EXEC all-ones within this wave).
#pragma unroll
  for (int m = 16; m >= 1; m >>= 1) {
    s  += __shfl_xor(s,  m, 32);
    sq += __shfl_xor(sq, m, 32);
  }
  const float mu  = s * (1.0f / H_DIM);
  const float var = sq * (1.0f / H_DIM) - mu * mu;
  const float inv = rsqrtf(var + 1e-12f);

  float* orow = out + (size_t)row * H_DIM;
#pragma unroll
  for (int c = 0; c < 4; ++c) {
    const int e = lane * 4 + c * 128;
    v4f o;
    o.x = (v[c * 4 + 0] - mu) * inv * swb[e + 0] + swb[H_DIM + e + 0];
    o.y = (v[c * 4 + 1] - mu) * inv * swb[e + 1] + swb[H_DIM + e + 1];
    o.z = (v[c * 4 + 2] - mu) * inv * swb[e + 2] + swb[H_DIM + e + 2];
    o.w = (v[c * 4 + 3] - mu) * inv * swb[e + 3] + swb[H_DIM + e + 3];
    // Streaming store with non-temporal hint (TH=NT): keep L2 for the tables.
    __builtin_nontemporal_store(o, (v4f*)(orow + e));
  }
}

// ---------------------------------------------------------------------------
extern "C" void kernel_launch(void* const* d_in, const int* in_sizes, int n_in,
                              void* d_out, int out_size, void* d_ws, size_t ws_size,
                              hipStream_t stream) {
  const int*   state  = (const int*)d_in[0];     // [B, S] int32
  const float* tok    = (const float*)d_in[1];   // [V, H]
  const float* ins    = (const float*)d_in[2];   // [P, H]
  const float* argt   = (const float*)d_in[3];   // [P, H]
  const float* w      = (const float*)d_in[4];   // [H]
  const float* bi     = (const float*)d_in[5];   // [H]
  const int*   nextId = (const int*)d_in[6];     // scalar
  float*       out    = (float*)d_out;           // [B, S, H] fp32

  const int nrows = in_sizes[0];                 // B*S
  const int B = nrows / S_LEN;

  int2* idxbuf = (int2*)d_ws;                    // nrows * 8 bytes

  scan_kernel<<<B, THREADS, 0, stream>>>(state, nextId, idxbuf);

  const int blocks = (nrows + WAVES_PER_BLOCK - 1) / WAVES_PER_BLOCK;
  emb_ln_kernel<<<blocks, THREADS, 0, stream>>>(state, idxbuf, tok, ins, argt,
                                                w, bi, out, nrows);
}